// NPCLoss_49924699848799
// MI455X (gfx1250) — compile-verified
//
#include <hip/hip_runtime.h>
#include <hip/hip_bf16.h>
#include <stdint.h>

// ---------------------------------------------------------------------------
// NPCLoss for MI455X (gfx1250).
// Phase 1: per-row fused top-2 + online logsumexp, streaming 1.05 GB once
//          through the CDNA5 async global->LDS engine (4-deep pipeline).
// Phase 2: single-block sort/scan/threshold over 8192 scalars.
// ---------------------------------------------------------------------------

__device__ __forceinline__ void async_ld_b128(uint32_t lds_addr, uint32_t voff,
                                              uint64_t sbase) {
  // GVS mode: mem_addr = SGPR64 + VGPR32 offset; VDST VGPR carries LDS byte addr.
  asm volatile("global_load_async_to_lds_b128 %0, %1, %2 th:TH_LOAD_NT"
               :: "v"(lds_addr), "v"(voff), "s"(sbase)
               : "memory");
}

// Branchless online (top2, logsumexp) update: exactly one v_exp_f32 per element.
#define ACC(xv)                                                   \
  do {                                                            \
    float _x = (xv);                                              \
    float _e = __expf(-fabsf(_x - m1));                           \
    bool  _g = (_x > m1);                                         \
    s  = fmaf(s, _g ? _e : 1.f, _g ? 1.f : _e);                   \
    m2 = fmaxf(m2, fminf(m1, _x));                                \
    m1 = fmaxf(m1, _x);                                           \
  } while (0)

__global__ __launch_bounds__(256) void npc_row_kernel(
    const float* __restrict__ logits, const int* __restrict__ target,
    float* __restrict__ loss_ws, float* __restrict__ margin_ws,
    int N, int C)
{
  constexpr int NB = 4;                   // async pipeline depth
  const int tid = threadIdx.x;
  const int row = blockIdx.x;
  if (row >= N) return;

  const float* rowp = logits + (size_t)row * (size_t)C;
  const uint64_t sbase = (uint64_t)(uintptr_t)rowp;

  __shared__ float4 buf[NB][256];         // 16 KB ring buffer
  __shared__ float r_m1[256], r_m2[256], r_s[256];

  // Low 32 bits of a flat LDS pointer == wave-relative LDS byte address.
  const uint32_t lds_base = (uint32_t)(uintptr_t)&buf[0][tid];

  float m1 = -INFINITY, m2 = -INFINITY, s = 0.f;

  const int nvec = C >> 2;                // float4 elements per row
  const int full = nvec >> 8;             // iterations where all 256 lanes load

  // ---- prologue: fill the pipeline ---------------------------------------
  for (int p = 0; p < NB && p < full; ++p)
    async_ld_b128(lds_base + ((uint32_t)p << 12),
                  (uint32_t)((p * 256 + tid) * 16), sbase);

  // ---- main pipelined loop ------------------------------------------------
  int it = 0;
  for (; it + NB < full; ++it) {
    asm volatile("s_wait_asynccnt 3" ::: "memory");   // oldest buffer landed
    float4 v = buf[it & (NB - 1)][tid];
    // DS read must complete before async engine may overwrite this buffer.
    asm volatile("s_wait_dscnt 0" ::: "memory");
    async_ld_b128(lds_base + ((uint32_t)((it + NB) & (NB - 1)) << 12),
                  (uint32_t)(((it + NB) * 256 + tid) * 16), sbase);
    ACC(v.x); ACC(v.y); ACC(v.z); ACC(v.w);
  }
  // ---- drain --------------------------------------------------------------
  for (; it < full; ++it) {
    int rem = full - 1 - it;
    if (rem >= 3)      asm volatile("s_wait_asynccnt 3" ::: "memory");
    else if (rem == 2) asm volatile("s_wait_asynccnt 2" ::: "memory");
    else if (rem == 1) asm volatile("s_wait_asynccnt 1" ::: "memory");
    else               asm volatile("s_wait_asynccnt 0" ::: "memory");
    float4 v = buf[it & (NB - 1)][tid];
    ACC(v.x); ACC(v.y); ACC(v.z); ACC(v.w);
  }

  // ---- vec4 tail (nvec % 256 lanes worth) ---------------------------------
  int vbase = full * 256;
  if (vbase + tid < nvec) {
    float4 v = ((const float4*)rowp)[vbase + tid];
    ACC(v.x); ACC(v.y); ACC(v.z); ACC(v.w);
  }
  // ---- scalar tail (C % 4) ------------------------------------------------
  int srem = C & 3;
  if (tid < srem) { ACC(rowp[(C & ~3) + tid]); }

  // ---- cross-thread reduction of (m1, m2, s) ------------------------------
  r_m1[tid] = m1; r_m2[tid] = m2; r_s[tid] = s;
  __syncthreads();
  for (int st = 128; st > 0; st >>= 1) {
    if (tid < st) {
      float a1 = r_m1[tid], a2 = r_m2[tid], as = r_s[tid];
      float b1 = r_m1[tid + st], b2 = r_m2[tid + st], bs = r_s[tid + st];
      float M1 = fmaxf(a1, b1);
      float M2 = fmaxf(fminf(a1, b1), fmaxf(a2, b2));
      float ea = (a1 == -INFINITY) ? 0.f : __expf(a1 - M1);
      float eb = (b1 == -INFINITY) ? 0.f : __expf(b1 - M1);
      r_m1[tid] = M1; r_m2[tid] = M2; r_s[tid] = as * ea + bs * eb;
    }
    __syncthreads();
  }

  if (tid == 0) {
    float M1 = r_m1[0], M2 = r_m2[0], S = r_s[0];
    float lse = M1 + __logf(S);
    int tgt = target[row];
    float tl = rowp[tgt];
    float margin1 = tl - M1;
    float margin2 = tl - M2;
    float margin  = (margin1 > 0.f) ? margin1 : margin2;
    float fst = fmaxf(0.f, 1.f - margin);
    float snd = fmaxf(0.f, 1.f - tl + lse);
    loss_ws[row]   = (margin >= 0.f) ? fst : snd;
    margin_ws[row] = margin;
  }
}

__global__ __launch_bounds__(1024) void npc_final_kernel(
    const float* __restrict__ loss_ws, const float* __restrict__ margin_ws,
    float* __restrict__ out, int N)
{
  constexpr int M = 8192;                 // pow2 sort size (>= N)
  constexpr int PT = M / 1024;            // 8 elements per thread
  const int tid = threadIdx.x;

  __shared__ float s_val[M];              // 32 KB
  __shared__ float s_csum[M];             // 32 KB
  __shared__ float s_aux[1024];
  __shared__ float s_red[1024];
  __shared__ float s_thr;

  // ---- load losses + count negative margins -------------------------------
  float cnt = 0.f;
  for (int i = tid; i < M; i += 1024) {
    if (i < N) {
      s_val[i] = loss_ws[i];
      if (margin_ws[i] < 0.f) cnt += 1.f;
    } else {
      s_val[i] = INFINITY;                // padding sorts to the end
    }
  }
  s_red[tid] = cnt;
  __syncthreads();
  for (int st = 512; st > 0; st >>= 1) {
    if (tid < st) s_red[tid] += s_red[tid + st];
    __syncthreads();
  }
  if (tid == 0) {
    float nf = (float)N;                  // EPSILON = 0.2
    s_thr = 0.64f * nf + 0.8f * s_red[0];
  }
  __syncthreads();

  // ---- bitonic sort ascending --------------------------------------------
  for (int k = 2; k <= M; k <<= 1) {
    for (int j = k >> 1; j > 0; j >>= 1) {
      for (int i = tid; i < M; i += 1024) {
        int ixj = i ^ j;
        if (ixj > i) {
          float a = s_val[i], b = s_val[ixj];
          bool up = ((i & k) == 0);
          if ((a > b) == up) { s_val[i] = b; s_val[ixj] = a; }
        }
      }
      __syncthreads();
    }
  }

  // ---- inclusive cumsum (blocked scan) ------------------------------------
  const int base = tid * PT;
  float loc[PT];
  float run = 0.f;
  #pragma unroll
  for (int j = 0; j < PT; ++j) { run += s_val[base + j]; loc[j] = run; }
  s_aux[tid] = run;
  __syncthreads();
  for (int off = 1; off < 1024; off <<= 1) {
    float v = s_aux[tid];
    float add = (tid >= off) ? s_aux[tid - off] : 0.f;
    __syncthreads();
    s_aux[tid] = v + add;
    __syncthreads();
  }
  float excl = (tid > 0) ? s_aux[tid - 1] : 0.f;
  #pragma unroll
  for (int j = 0; j < PT; ++j) s_csum[base + j] = loc[j] + excl;
  __syncthreads();

  // ---- threshold keep + two masked sums -----------------------------------
  float thr = s_thr;
  float p1 = 0.f, p2 = 0.f;
  #pragma unroll
  for (int j = 0; j < PT; ++j) {
    int i = base + j;
    float v = s_val[i];
    if (s_csum[i] <= thr - (float)i) {
      p1 += v;
      p2 += (v + 1e-10f) / v;             // Binary(): matches reference (inf at v==0)
    }
  }
  s_red[tid] = p1; s_aux[tid] = p2;
  __syncthreads();
  for (int st = 512; st > 0; st >>= 1) {
    if (tid < st) { s_red[tid] += s_red[tid + st]; s_aux[tid] += s_aux[tid + st]; }
    __syncthreads();
  }
  if (tid == 0) {
    float npcl1 = s_red[0];
    float npcl2 = thr - s_aux[0];
    out[0] = fmaxf(npcl1, npcl2) / (float)N * 0.1f;
  }
}

extern "C" void kernel_launch(void* const* d_in, const int* in_sizes, int n_in,
                              void* d_out, int out_size, void* d_ws, size_t ws_size,
                              hipStream_t stream) {
  const float* logits = (const float*)d_in[0];   // [N, C] fp32
  const int*   target = (const int*)d_in[1];     // [N] int
  int N = in_sizes[1];
  int C = in_sizes[0] / N;

  float* loss_ws   = (float*)d_ws;               // [N]
  float* margin_ws = loss_ws + N;                 // [N]

  npc_row_kernel<<<N, 256, 0, stream>>>(logits, target, loss_ws, margin_ws, N, C);
  npc_final_kernel<<<1, 1024, 0, stream>>>(loss_ws, margin_ws, (float*)d_out, N);
}